// Cross_30932354466025
// MI455X (gfx1250) — compile-verified
//
#include <hip/hip_runtime.h>
#include <cstdint>

// out[b,d] = x0[b,d] * (x_cross[b,:] . w) + bias[d] + x_cross[b,d]
// B=16384, D=4096, f32. HBM-bound: ~768 MiB moved, ~0.33 GFLOP.
// Strategy: 1 row per block; async-stage x_cross row into LDS once
// (CDNA5 GLOBAL_LOAD_ASYNC_TO_LDS_B128 / ASYNCcnt), consume twice from LDS;
// stream x0/out with non-temporal B128; keep w/bias temporal (L2-resident).

#define BATCH_N 16384
#define D_N     4096
#define TPB     256            // 8 wave32s
#define VPT     4              // 4 x float4 per thread = 16 floats; 256*16 = 4096

typedef float v4f __attribute__((ext_vector_type(4)));

__global__ __launch_bounds__(TPB) void cross_fused_kernel(
    const float* __restrict__ x0,
    const float* __restrict__ xc,
    const float* __restrict__ w,
    const float* __restrict__ bias,
    float* __restrict__ out)
{
    __shared__ float lds_xc[D_N];     // 16 KiB staging for this row of x_cross
    __shared__ float red[TPB / 32];   // one partial per wave32
    __shared__ float s_bcast;

    const int tid = threadIdx.x;
    const int row = blockIdx.x;

    const uint32_t row_byte = (uint32_t)row * (uint32_t)(D_N * 4u);
    // Low 32 bits of the flat address of a __shared__ object == LDS byte offset
    // (LDS aperture maps addr[31:0] -> wave LDS allocation).
    const uint32_t lds_base = (uint32_t)(uintptr_t)&lds_xc[0];
    const uint64_t xc_base  = (uint64_t)(uintptr_t)xc;

    // ---- Phase 0: async-stage x_cross row into LDS (ASYNCcnt path) ----
    // GVS mode: mem_addr = SGPR64 + VGPR32_offset; per-lane LDS dest in VGPR.
    // Non-temporal: each row is touched by exactly one block.
#pragma unroll
    for (int i = 0; i < VPT; ++i) {
        const uint32_t chunk    = (uint32_t)(i * TPB + tid) * 16u; // byte offset in row
        const uint32_t lds_addr = lds_base + chunk;
        const uint32_t voff     = row_byte + chunk;
        asm volatile(
            "global_load_async_to_lds_b128 %0, %1, %2 th:TH_LOAD_NT"
            :
            : "v"(lds_addr), "v"(voff), "s"(xc_base)
            : "memory");
    }
    // Drain this wave's async-to-LDS transfers before reading them back.
    asm volatile("s_wait_asynccnt 0" ::: "memory");

    // ---- Phase 1: partial dot  x_cross[row,:] . w  ----
    float partial = 0.0f;
#pragma unroll
    for (int i = 0; i < VPT; ++i) {
        const int e = (i * TPB + tid) * 4;                 // float index in row
        const v4f xv = *reinterpret_cast<const v4f*>(&lds_xc[e]); // ds_load_b128
        const v4f wv = *reinterpret_cast<const v4f*>(&w[e]);      // temporal, L2-hot
        partial = fmaf(xv.x, wv.x, partial);
        partial = fmaf(xv.y, wv.y, partial);
        partial = fmaf(xv.z, wv.z, partial);
        partial = fmaf(xv.w, wv.w, partial);
    }

    // wave32 butterfly reduction (warpSize == 32 on gfx1250)
#pragma unroll
    for (int off = 16; off > 0; off >>= 1)
        partial += __shfl_xor(partial, off, 32);

    if ((tid & 31) == 0) red[tid >> 5] = partial;
    __syncthreads();
    if (tid == 0) {
        float s = 0.0f;
#pragma unroll
        for (int k = 0; k < TPB / 32; ++k) s += red[k];
        s_bcast = s;
    }
    __syncthreads();
    const float s = s_bcast;

    // ---- Phase 2: out = x0 * s + bias + x_cross ----
    const size_t row_off = (size_t)row * (size_t)D_N;
#pragma unroll
    for (int i = 0; i < VPT; ++i) {
        const int e = (i * TPB + tid) * 4;
        const v4f xv = *reinterpret_cast<const v4f*>(&lds_xc[e]);   // from LDS again
        const v4f bv = *reinterpret_cast<const v4f*>(&bias[e]);     // temporal
        const v4f x0v = __builtin_nontemporal_load(
            reinterpret_cast<const v4f*>(x0 + row_off + e));        // streaming NT

        v4f o;
        o.x = fmaf(x0v.x, s, bv.x + xv.x);
        o.y = fmaf(x0v.y, s, bv.y + xv.y);
        o.z = fmaf(x0v.z, s, bv.z + xv.z);
        o.w = fmaf(x0v.w, s, bv.w + xv.w);

        __builtin_nontemporal_store(o, reinterpret_cast<v4f*>(out + row_off + e));
    }
}

extern "C" void kernel_launch(void* const* d_in, const int* in_sizes, int n_in,
                              void* d_out, int out_size, void* d_ws, size_t ws_size,
                              hipStream_t stream) {
    const float* x0  = (const float*)d_in[0];
    const float* xc  = (const float*)d_in[1];
    const float* w   = (const float*)d_in[2];
    const float* b   = (const float*)d_in[3];
    float* out       = (float*)d_out;

    dim3 grid(BATCH_N);
    dim3 block(TPB);
    cross_fused_kernel<<<grid, block, 0, stream>>>(x0, xc, w, b, out);
}